// SAGEConv_16604343566549
// MI455X (gfx1250) — compile-verified
//
#include <hip/hip_runtime.h>

typedef __attribute__((ext_vector_type(2))) float v2f;
typedef __attribute__((ext_vector_type(8))) float v8f;

#define D 128
#define LDSTRIDE 132   // padded row stride (dwords): conflict-free ds_load_b64

// ---------------------------------------------------------------------------
// Kernel 1: zero agg (N*D) and cnt (N)
// ---------------------------------------------------------------------------
__global__ void sage_zero_kernel(float* __restrict__ buf, long long total) {
    long long i = (long long)blockIdx.x * blockDim.x + threadIdx.x;
    long long stride = (long long)gridDim.x * blockDim.x;
    for (; i < total; i += stride) buf[i] = 0.0f;
}

// ---------------------------------------------------------------------------
// Kernel 2: edge scatter. One wave (32 lanes) per edge: coalesced 512B row
// read of x[src], per-lane float4 atomic-add into agg[dst], one cnt atomic.
// x (51MB) and agg (51MB) are L2-resident (192MB L2) -> L2 atomic bound.
// ---------------------------------------------------------------------------
__global__ void sage_scatter_kernel(const long long* __restrict__ ei,
                                    const float* __restrict__ x,
                                    float* __restrict__ agg,
                                    float* __restrict__ cnt,
                                    int E) {
    int gtid = blockIdx.x * blockDim.x + threadIdx.x;
    int wave = gtid >> 5;
    int lane = threadIdx.x & 31;
    if (wave >= E) return;

    int src = (int)ei[wave];                 // row 0 of edge_index
    int dst = (int)ei[(long long)E + wave];  // row 1 of edge_index

    const float4* xr = (const float4*)(x + (long long)src * D);
    float4 v = xr[lane];                     // coalesced: 32 lanes * 16B = 512B

    float* ar = agg + (long long)dst * D + lane * 4;
    unsafeAtomicAdd(ar + 0, v.x);            // global_atomic_add_f32
    unsafeAtomicAdd(ar + 1, v.y);
    unsafeAtomicAdd(ar + 2, v.z);
    unsafeAtomicAdd(ar + 3, v.w);
    if (lane == 0) unsafeAtomicAdd(cnt + dst, 1.0f);
}

// ---------------------------------------------------------------------------
// Kernel 3: out = (agg/max(cnt,1)) @ W_l^T + x @ W_r^T with
// V_WMMA_F32_16X16X4_F32. Block = 256 threads (8 waves); the block stages
// both 128x128 weight matrices into LDS (padded stride 132 -> conflict-free
// b64 reads), then each wave computes one 16-row x 128-col output strip:
// 8 f32 accumulator tiles in VGPRs, K looped in steps of 4.
//
// Fragment layouts (ISA 7.12.2, wave32):
//   A 16x4 f32 : lanes L / L+16 hold row M=L; VGPR0 = K={0 | 2}, VGPR1 = K={1 | 3}
//   B 4x16 f32 : lanes 0-15 = N=0..15 (K=ka in .x, ka+1 in .y), symmetric
//   C/D        : VGPR v -> M = v + 8*hi, N = lane%16
// ---------------------------------------------------------------------------
__global__ void sage_gemm_kernel(const float* __restrict__ agg,
                                 const float* __restrict__ cnt,
                                 const float* __restrict__ x,
                                 const float* __restrict__ Wl,
                                 const float* __restrict__ Wr,
                                 float* __restrict__ out,
                                 int N) {
    extern __shared__ float lds[];
    float* lWl = lds;                        // [128][LDSTRIDE]
    float* lWr = lds + D * LDSTRIDE;

    int tid = threadIdx.x;

    // ---- stage Wl, Wr into LDS: float4-coalesced global reads, 16B-aligned
    // conflict-free LDS stores (row stride 132 dwords, 132 % 4 == 0) ----
#pragma unroll
    for (int j = 0; j < 16; ++j) {
        int e = (j * 256 + tid) * 4;         // linear element index, 16384/mat
        int n = e >> 7;                      // row (output feature)
        int k = e & 127;                     // col (input feature)
        *(float4*)&lWl[n * LDSTRIDE + k] = *(const float4*)(Wl + e);
        *(float4*)&lWr[n * LDSTRIDE + k] = *(const float4*)(Wr + e);
    }
    __syncthreads();                         // ALL threads reach this barrier

    int gtid = blockIdx.x * blockDim.x + tid;
    int wave = gtid >> 5;
    int base = wave * 16;
    if (base < N) {                          // wave-uniform: EXEC all-1s inside
        int lane = tid & 31;
        int col  = lane & 15;
        int hi   = lane >> 4;

        int gr = base + col;
        if (gr > N - 1) gr = N - 1;          // clamp reads (stores guarded)

        float c    = cnt[gr];
        float rinv = 1.0f / fmaxf(c, 1.0f);

        const float* aggRow = agg + (long long)gr * D;
        const float* xRow   = x   + (long long)gr * D;

        v8f acc[8];
#pragma unroll
        for (int t = 0; t < 8; ++t)
            acc[t] = (v8f){0.f,0.f,0.f,0.f,0.f,0.f,0.f,0.f};

#pragma unroll 2
        for (int k0 = 0; k0 < D; k0 += 4) {
            int ka = k0 + 2 * hi;

            v2f aM = *(const v2f*)(aggRow + ka);
            aM.x *= rinv; aM.y *= rinv;      // fuse mean = agg / max(cnt,1)
            v2f aX = *(const v2f*)(xRow + ka);

#pragma unroll
            for (int t = 0; t < 8; ++t) {
                // B[k][n] = W[n][k]; ds_load_b64, banks (4*col+ka)%64 -> no conflict
                v2f bl = *(const v2f*)&lWl[(t * 16 + col) * LDSTRIDE + ka];
                v2f br = *(const v2f*)&lWr[(t * 16 + col) * LDSTRIDE + ka];
                acc[t] = __builtin_amdgcn_wmma_f32_16x16x4_f32(
                    false, aM, false, bl, (short)0, acc[t], false, false);
                acc[t] = __builtin_amdgcn_wmma_f32_16x16x4_f32(
                    false, aX, false, br, (short)0, acc[t], false, false);
            }
        }

#pragma unroll
        for (int t = 0; t < 8; ++t) {
#pragma unroll
            for (int v = 0; v < 8; ++v) {
                int r = base + v + 8 * hi;
                if (r < N) out[(long long)r * D + t * 16 + col] = acc[t][v];
            }
        }
    }
}

// ---------------------------------------------------------------------------
// Launch
// ---------------------------------------------------------------------------
extern "C" void kernel_launch(void* const* d_in, const int* in_sizes, int n_in,
                              void* d_out, int out_size, void* d_ws, size_t ws_size,
                              hipStream_t stream) {
    const float*     x  = (const float*)d_in[0];
    const long long* ei = (const long long*)d_in[1];   // int64 edge_index [2, E]
    const float*     Wl = (const float*)d_in[2];
    const float*     Wr = (const float*)d_in[3];
    float* out = (float*)d_out;

    int N = in_sizes[0] / D;
    int E = in_sizes[1] / 2;

    float* agg = (float*)d_ws;                 // N*D floats
    float* cnt = agg + (long long)N * D;       // N floats

    // 1) zero accumulators (deterministic per call)
    long long total = (long long)N * D + N;
    int zb = (int)((total + 255) / 256);
    if (zb > 65535) zb = 65535;
    sage_zero_kernel<<<zb, 256, 0, stream>>>(agg, total);

    // 2) edge scatter: 8 edges per 256-thread block (one wave each)
    int sb = (E + 7) / 8;
    sage_scatter_kernel<<<sb, 256, 0, stream>>>(ei, x, agg, cnt, E);

    // 3) fused mean + dual GEMM via WMMA: 8 row-strips per block,
    //    weights staged in LDS (2 * 128 * 132 * 4 B = 135168 B <= 320KB/WG)
    int strips = (N + 15) / 16;
    int gb = (strips + 7) / 8;
    size_t ldsBytes = (size_t)2 * D * LDSTRIDE * sizeof(float);
    sage_gemm_kernel<<<gb, 256, ldsBytes, stream>>>(agg, cnt, x, Wl, Wr, out, N);
}